// CBOWModel_78305843741043
// MI455X (gfx1250) — compile-verified
//
#include <hip/hip_runtime.h>
#include <hip/hip_bf16.h>

// CBOW negative-sampling loss, fused single pass + tiny deterministic reduce.
// Memory-bound: ~610 MB of HBM gathers -> ~26us floor at 23.3 TB/s.
// Scores computed with V_WMMA_F32_16X16X4_F32 (full f32), one batch element
// per wave32, 16 out-rows per WMMA tile, K accumulated in chunks of 4.

typedef float v8f __attribute__((ext_vector_type(8)));
typedef float v4f __attribute__((ext_vector_type(4)));
typedef float v2f __attribute__((ext_vector_type(2)));

#define D_DIM 300
#define NCTX_C 10
#define NNEG_C 20
#define WAVES_PER_BLOCK 8
#define CM_STRIDE 304   // floats per wave (padded, keeps 16B alignment: 304*4=1216)

__device__ __forceinline__ float log_sigmoid(float x) {
    // stable: min(x,0) - log1p(exp(-|x|))
    float ax = __builtin_fabsf(x);
    return __builtin_fminf(x, 0.0f) - log1pf(__expf(-ax));
}

__global__ __launch_bounds__(256) void cbow_loss_kernel(
    const int*   __restrict__ ctx,        // [B, NCTX]
    const int*   __restrict__ center,     // [B]
    const int*   __restrict__ neg,        // [B, NNEG]
    const float* __restrict__ in_embed,   // [V, D]
    const float* __restrict__ out_embed,  // [V, D]
    float*       __restrict__ block_partials,  // [gridDim.x]
    int Btot)
{
    __shared__ float cm_all[WAVES_PER_BLOCK * CM_STRIDE];
    __shared__ float wsum[WAVES_PER_BLOCK];

    const int lane = threadIdx.x & 31;
    const int wv   = threadIdx.x >> 5;
    const int b    = blockIdx.x * WAVES_PER_BLOCK + wv;   // wave-uniform
    const int hi   = lane >> 4;                           // half-wave id (K lane group)

    float tot = 0.0f;
    if (b < Btot) {
        const int crow = center[b];

        // Prefetch the 21 out_embed rows we will stream in phase 2.
        {
            int m    = lane;
            int prow = (m >= 1 && m <= NNEG_C) ? neg[b * NNEG_C + (m - 1)] : crow;
            __builtin_prefetch(out_embed + (size_t)prow * D_DIM, 0, 3);
        }

        // ---------------- Phase 1: mean-pool 10 context rows (coalesced b128) ----
        // Row base = row*1200 bytes, 16B-aligned; lane owns floats [4L,4L+3] etc.
        v4f acc0 = {0.f, 0.f, 0.f, 0.f};
        v4f acc1 = {0.f, 0.f, 0.f, 0.f};
        v4f acc2 = {0.f, 0.f, 0.f, 0.f};
        #pragma unroll
        for (int j = 0; j < NCTX_C; ++j) {
            const int row = ctx[b * NCTX_C + j];          // wave-uniform -> SMEM
            const v4f* p  = (const v4f*)(in_embed + (size_t)row * D_DIM);
            acc0 += p[lane];                              // floats   0..127
            acc1 += p[lane + 32];                         // floats 128..255
            if (lane < 11) acc2 += p[lane + 64];          // floats 256..299
        }
        const float im = 1.0f / (float)NCTX_C;
        float* cm = cm_all + wv * CM_STRIDE;              // wave-private ctx mean
        v4f*  cmv = (v4f*)cm;
        cmv[lane]      = acc0 * im;
        cmv[lane + 32] = acc1 * im;
        if (lane < 11) cmv[lane + 64] = acc2 * im;
        // Same-wave DS store->load is hardware in-order; no block barrier needed.

        // ---------------- Phase 2: 21 dot products via V_WMMA_F32_16X16X4_F32 ----
        // A (16x4, M=out-row, K=feature chunk): lane L -> row M=(L&15),
        //   v0/v1 = E[M][4kc + 2*hi + {0,1}]  (per ISA 32-bit A layout).
        // B (4x16): ctx-mean chunk broadcast across all N columns, so every
        //   output column equals the score vector (robust to N striping).
        // D (16x16): reg r holds M = r + 8*hi, identical across the 16 N-lanes.
        const v2f* cm2 = (const v2f*)cm;
        #pragma unroll
        for (int t = 0; t < 2; ++t) {
            const int m   = (lane & 15) + t * 16;         // out-row this lane feeds
            const int row = (m >= 1 && m <= NNEG_C) ? neg[b * NNEG_C + (m - 1)]
                                                    : crow;  // m==0: center; pads: dup
            const v2f* ep2 = (const v2f*)(out_embed + (size_t)row * D_DIM);

            v8f d = {0.f, 0.f, 0.f, 0.f, 0.f, 0.f, 0.f, 0.f};
            #pragma unroll 5
            for (int kc = 0; kc < D_DIM / 4; ++kc) {      // 75 K-chunks of 4
                v2f av = ep2[kc * 2 + hi];                // global, 8B, streams the row
                v2f bv = cm2[kc * 2 + hi];                // LDS broadcast (2 addrs/wave)
                d = __builtin_amdgcn_wmma_f32_16x16x4_f32(
                        false, av, false, bv, (short)0, d, false, false);
            }

            // Harvest scores: reg r -> M = t*16 + r + 8*hi (same on all 16 N-lanes).
            #pragma unroll
            for (int r = 0; r < 8; ++r) {
                const int mm = t * 16 + r + hi * 8;
                const float s = d[r];
                if (mm == 0)            tot += log_sigmoid(s);    // positive score
                else if (mm <= NNEG_C)  tot += log_sigmoid(-s);   // negatives
            }
        }
        // lanes 0-15 hold the low-row partial, 16-31 the high-row partial.
        tot += __shfl_xor(tot, 16, 32);
    }

    if (lane == 0) wsum[wv] = tot;
    __syncthreads();
    if (threadIdx.x == 0) {
        float s = 0.0f;
        #pragma unroll
        for (int i = 0; i < WAVES_PER_BLOCK; ++i) s += wsum[i];
        block_partials[blockIdx.x] = s;
    }
}

__global__ __launch_bounds__(256) void cbow_reduce_kernel(
    const float* __restrict__ partials, float* __restrict__ out,
    int nb, float invB)
{
    __shared__ float sm[256];
    float s = 0.0f;
    for (int i = threadIdx.x; i < nb; i += 256) s += partials[i];
    sm[threadIdx.x] = s;
    __syncthreads();
    for (int st = 128; st > 0; st >>= 1) {
        if (threadIdx.x < st) sm[threadIdx.x] += sm[threadIdx.x + st];
        __syncthreads();
    }
    if (threadIdx.x == 0) out[0] = -sm[0] * invB;   // loss = -mean(total_b)
}

extern "C" void kernel_launch(void* const* d_in, const int* in_sizes, int n_in,
                              void* d_out, int out_size, void* d_ws, size_t ws_size,
                              hipStream_t stream) {
    const int*   ctx      = (const int*)  d_in[0];   // [B,10]
    const int*   center   = (const int*)  d_in[1];   // [B]
    const int*   neg      = (const int*)  d_in[2];   // [B,20]
    const float* in_w     = (const float*)d_in[3];   // [V,300]
    const float* out_w    = (const float*)d_in[4];   // [V,300]
    float*       out      = (float*)d_out;
    float*       partials = (float*)d_ws;            // nblocks floats (~8KB)

    const int B       = in_sizes[1];
    const int nblocks = (B + WAVES_PER_BLOCK - 1) / WAVES_PER_BLOCK;

    cbow_loss_kernel<<<nblocks, 256, 0, stream>>>(ctx, center, neg, in_w, out_w,
                                                  partials, B);
    cbow_reduce_kernel<<<1, 256, 0, stream>>>(partials, out, nblocks, 1.0f / (float)B);
}